// YINTransform_13967233647310
// MI455X (gfx1250) — compile-verified
//
#include <hip/hip_runtime.h>
#include <hip/hip_bf16.h>
#include <math.h>

typedef __attribute__((ext_vector_type(2))) float v2f;
typedef __attribute__((ext_vector_type(8))) float v8f;

#define WINLEN 1480
#define NFR    499
#define HOPSZ  320
#define TSIG   160000
#define NTILE  93          // ceil(1480/16) lag tiles
#define LDSF   2048        // padded frame (worst fragment read index = 1983)
#define NMIDI  80

__global__ __launch_bounds__(128)
void yin_wmma_kernel(const float* __restrict__ x, float* __restrict__ out) {
  __shared__ float F[LDSF];            // zero-padded frame
  __shared__ float csum[WINLEN + 8];   // prefix sum of squares, [0..1480]
  __shared__ float acf[WINLEN + 8];    // accumulated autocorrelation (+guard lags)
  __shared__ float dfs[WINLEN];        // YIN difference function
  __shared__ float cmndf[WINLEN];      // normalized difference
  __shared__ float scr[4][256];        // per-wave 16x16 tile spill

  const int tid  = threadIdx.x;
  const int wave = tid >> 5;
  const int lane = tid & 31;
  const int half = lane >> 4;          // 0: lanes 0-15, 1: lanes 16-31
  const int m16  = lane & 15;

  const int fr = blockIdx.x % NFR;
  const int b  = blockIdx.x / NFR;
  const long base = (long)b * TSIG + (long)fr * HOPSZ;

  // valid float count for this frame (always a multiple of 4 -> 16B aligned)
  const int V = min(WINLEN, TSIG - fr * HOPSZ);

  // ---- zero pad region [V, LDSF) with plain DS stores (disjoint from async) ----
  for (int t = V + tid; t < LDSF; t += 128) F[t] = 0.0f;
  for (int i = tid; i < WINLEN + 8; i += 128) acf[i] = 0.0f;

  // ---- async DMA frame into LDS: global_load_async_to_lds_b128 ----
  for (int r = 0; r < 3; ++r) {
    int t = (r * 128 + tid) * 4;                 // 4 floats (16B) per lane
    if (t < V) {
      const float* gp = x + base + t;
      unsigned lds = (unsigned)(unsigned long long)(const void*)&F[t];
      asm volatile("global_load_async_to_lds_b128 %0, %1, off"
                   :: "v"(lds), "v"(gp) : "memory");
    }
  }
  asm volatile("s_wait_asynccnt 0x0" ::: "memory");
  __syncthreads();

  // ---- wave 0: inclusive shuffle-scan of F^2 -> csum ----
  if (wave == 0) {
    if (lane == 0) csum[0] = 0.0f;
    float carry = 0.0f;
    for (int c = 0; c < 47; ++c) {
      int t = c * 32 + lane;
      float v = (t < WINLEN) ? F[t] * F[t] : 0.0f;
#pragma unroll
      for (int off = 1; off < 32; off <<= 1) {
        float n = __shfl_up(v, off, 32);
        if (lane >= off) v += n;
      }
      float tot = carry + v;
      if (t < WINLEN) csum[t + 1] = tot;
      carry = __shfl(tot, 31, 32);
    }
  }

  // ---- WMMA autocorrelation: Toeplitz tiles, full utilization ----
  // Tile tau0: D[i][j] = sum_k F[16k+i]*F[16k+j+tau0] (residue-i partial of lag
  // tau0+j-i).  acf[tau0+d] = sum_i D_tau0[i][i+d] + sum_j D_{tau0+16}[j+16-d][j].
  float* myscr = scr[wave];
  const int aoff = (half << 1) * 16 + m16;       // 16*(2*half) + m16

  // Preload all A fragments into VGPRs (constant indices, shared by all tiles)
  v2f afr[24];
#pragma unroll
  for (int s = 0; s < 24; ++s) {
    int o = (s << 6) + aoff;
    afr[s].x = F[o];
    afr[s].y = F[o + 16];
  }

  const int wv = __builtin_amdgcn_readfirstlane(wave);   // scalar loop control
  for (int tile = wv; tile < NTILE; tile += 4) {
    const int tau0  = tile << 4;
    const int nb    = (1495 - tau0) >> 4;                // k-blocks with data
    const int steps = (nb + 3) >> 2;                     // true WMMA count, 1..24
    const int bbase = aoff + tau0;
    v8f c = {};
#pragma unroll
    for (int ch = 0; ch < 3; ++ch) {                     // 3 predicated chunks of 8
      if (ch * 8 < steps) {
#pragma unroll
        for (int u = 0; u < 8; ++u) {
          const int s = ch * 8 + u;                      // compile-time constant
          int o = (s << 6) + bbase;
          v2f bm;
          bm.x = F[o];
          bm.y = F[o + 16];
          c = __builtin_amdgcn_wmma_f32_16x16x4_f32(false, afr[s], false, bm,
                                                    (short)0, c, false, false);
        }
      }
    }
    // spill tile: D[M][N], M = v + 8*half, N = m16 (same-wave LDS is in-order)
#pragma unroll
    for (int v = 0; v < 8; ++v)
      myscr[(v + (half << 3)) * 16 + m16] = c[v];
    // fold diagonals into acf (two tiles feed each lag -> ds_add_f32 atomics)
    if (lane < 16) {
      int d = lane;
      float s0 = 0.0f;
      for (int i = 0; i + d < 16; ++i) s0 += myscr[i * 16 + i + d];
      atomicAdd(&acf[tau0 + d], s0);
    } else if (lane < 31) {
      int d = lane - 15;                                 // 1..15 -> lag tau0-16+d
      int lagv = tau0 - 16 + d;
      if (lagv >= 0) {
        float s0 = 0.0f;
        for (int j = 0; j < d; ++j) s0 += myscr[(j + 16 - d) * 16 + j];
        atomicAdd(&acf[lagv], s0);
      }
    }
  }
  __syncthreads();

  // ---- difference function ----
  for (int t = tid; t < WINLEN; t += 128)
    dfs[t] = csum[WINLEN - t] + csum[WINLEN] - csum[t] - 2.0f * acf[t];
  __syncthreads();

  // ---- wave 0: running-sum scan of df -> cmndf ----
  if (wave == 0) {
    if (lane == 0) cmndf[0] = 1.0f;
    float carry = 0.0f;
    for (int c = 0; c < 47; ++c) {
      int t = 1 + c * 32 + lane;
      float dv = (t < WINLEN) ? dfs[t] : 0.0f;
      float v = dv;
#pragma unroll
      for (int off = 1; off < 32; off <<= 1) {
        float n = __shfl_up(v, off, 32);
        if (lane >= off) v += n;
      }
      float tot = carry + v;
      if (t < WINLEN) cmndf[t] = dv * (float)t / tot;
      carry = __shfl(tot, 31, 32);
    }
  }
  __syncthreads();

  // ---- interpolate at MIDI lags, write (B, 80, 499) ----
  if (tid < NMIDI) {
    int m = 5 + tid;
    double lag = 16000.0 / (440.0 * exp2((double)(m - 69) / 12.0));
    int lf = (int)floor(lag);
    float w = (float)(lag - (double)lf);                 // lc = lf+1 (non-integer)
    float ylo = cmndf[lf];
    float yhi = cmndf[lf + 1];
    out[((long)b * NMIDI + tid) * NFR + fr] = (yhi - ylo) * w + ylo;
  }
}

extern "C" void kernel_launch(void* const* d_in, const int* in_sizes, int n_in,
                              void* d_out, int out_size, void* d_ws, size_t ws_size,
                              hipStream_t stream) {
  (void)in_sizes; (void)n_in; (void)d_ws; (void)ws_size; (void)out_size;
  const float* x = (const float*)d_in[0];
  float* out = (float*)d_out;
  yin_wmma_kernel<<<dim3(16 * NFR), dim3(128), 0, stream>>>(x, out);
}